// MyNet_33285996544616
// MI455X (gfx1250) — compile-verified
//
#include <hip/hip_runtime.h>
#include <math.h>

#define NN   17186      // nodes per graph
#define FIN  6
#define H1D  64
#define H2D  32
#define KK   12031      // kept nodes per graph (ceil(0.7*NN))
#define TILES 752       // ceil(KK/16)
#define TPB   8         // tiles per block in the WMMA kernel
#define GXW   94        // ceil(TILES/TPB)

typedef float v2f __attribute__((ext_vector_type(2)));
typedef float v8f __attribute__((ext_vector_type(8)));

// ---------------------------------------------------------------- init
__global__ void k_init(float* dinv1, float* acc1, float* sacc, int* nidx0,
                       float* dinv2, float* acc2, int* gbits, int* keep,
                       int Bsz, long long ntot) {
  long long i = (long long)blockIdx.x * blockDim.x + threadIdx.x;
  if (i >= ntot) return;
  if (i < (long long)NN * H1D) acc1[i] = 0.f;
  if (i < NN) { dinv1[i] = 1.f; sacc[i] = 0.f; nidx0[i] = -1; }
  if (i < KK) dinv2[i] = 1.f;
  if (i < (long long)KK * H2D) acc2[i] = 0.f;
  if (i < (long long)Bsz * H2D) gbits[i] = 0;
  if (i < (long long)Bsz * KK) keep[i] = (int)(i / KK) * NN;  // safe fallback node
}

// ------------------------------------------------------- conv1 degrees
__global__ void k_deg1(const int* dst, float* deg, int E) {
  int e = blockIdx.x * blockDim.x + threadIdx.x;
  if (e < E) atomicAdd(&deg[dst[e]], 1.f);
}
__global__ void k_rsqrt(float* v, int n) {
  int i = blockIdx.x * blockDim.x + threadIdx.x;
  if (i < n) v[i] = rsqrtf(v[i]);
}

// ------------------------------------------- conv1: xw1 for graph0 only
__global__ void k_xw1(const float* x, const float* W1, float* xh1) {
  int idx = blockIdx.x * blockDim.x + threadIdx.x;
  if (idx >= NN * H1D) return;
  int i = idx >> 6, f = idx & 63;
  const float* xr = x + (size_t)i * FIN;
  float t = 0.f;
#pragma unroll
  for (int c = 0; c < FIN; ++c) t += xr[c] * W1[c * H1D + f];
  xh1[idx] = t;
}

__global__ void k_conv1_edges(const int* src, const int* dst, const float* dinv1,
                              const float* xw1, float* acc1, int E) {
  int idx = blockIdx.x * blockDim.x + threadIdx.x;
  if (idx >= E * H1D) return;
  int e = idx >> 6, f = idx & 63;
  int s = src[e], d = dst[e];
  float w = dinv1[s] * dinv1[d];
  atomicAdd(&acc1[(size_t)d * H1D + f], xw1[(size_t)s * H1D + f] * w);
}

__global__ void k_h1fin(float* xh1, const float* acc1, const float* dinv1,
                        const float* b1) {
  int idx = blockIdx.x * blockDim.x + threadIdx.x;
  if (idx >= NN * H1D) return;
  int i = idx >> 6, f = idx & 63;
  float dv = dinv1[i];
  float v = acc1[idx] + xh1[idx] * dv * dv + b1[f];
  xh1[idx] = fmaxf(v, 0.f);   // in-place: xw1 -> h1(graph0)
}

// ------------------------------------------------------------- scoring
__global__ __launch_bounds__(256)
void k_score(const float* x, const float* xh1, const float* W1, const float* b1,
             const float* Wp, const float* bp, float* sxw0, float* score,
             long long Ntot) {
  __shared__ float sW1[FIN * H1D];
  __shared__ float sB1[H1D];
  __shared__ float sWp[H1D];
  int tid = threadIdx.x;
  for (int i = tid; i < FIN * H1D; i += 256) sW1[i] = W1[i];
  if (tid < H1D) { sB1[tid] = b1[tid]; sWp[tid] = Wp[tid]; }
  __syncthreads();
  long long i = (long long)blockIdx.x * blockDim.x + tid;
  if (i >= Ntot) return;
  if (i < NN) {                     // graph 0: pre-aggregation dot, finalized later
    const float* h = xh1 + i * H1D;
    float t = 0.f;
#pragma unroll
    for (int f = 0; f < H1D; ++f) t += h[f] * sWp[f];
    sxw0[i] = t;
  } else {                          // graphs >= 1: h1 recomputed on the fly
    const float* xr = x + i * FIN;
    float x0 = xr[0], x1 = xr[1], x2 = xr[2], x3 = xr[3], x4 = xr[4], x5 = xr[5];
    float t = 0.f;
#pragma unroll
    for (int f = 0; f < H1D; ++f) {
      float h = sB1[f] + x0 * sW1[f] + x1 * sW1[H1D + f] + x2 * sW1[2 * H1D + f]
                       + x3 * sW1[3 * H1D + f] + x4 * sW1[4 * H1D + f]
                       + x5 * sW1[5 * H1D + f];
      t += fmaxf(h, 0.f) * sWp[f];
    }
    score[i] = t + bp[0];
  }
}

__global__ void k_score_edges(const int* src, const int* dst, const float* dinv1,
                              const float* sxw0, float* sacc, int E) {
  int e = blockIdx.x * blockDim.x + threadIdx.x;
  if (e >= E) return;
  int s = src[e], d = dst[e];
  atomicAdd(&sacc[d], sxw0[s] * dinv1[s] * dinv1[d]);
}

__global__ void k_score0(const float* sacc, const float* sxw0, const float* dinv1,
                         const float* bp, float* score) {
  int i = blockIdx.x * blockDim.x + threadIdx.x;
  if (i >= NN) return;
  float dv = dinv1[i];
  score[i] = sacc[i] + sxw0[i] * dv * dv + bp[0];
}

// ----------------------------------------------- per-graph exact top-K
__global__ __launch_bounds__(256)
void k_topk(const float* score, int* keep, int* nidx0) {
  int g = blockIdx.x;
  const float* s = score + (size_t)g * NN;
  __shared__ int hist[256];
  __shared__ int scn[256];
  __shared__ unsigned sh_prefix;
  __shared__ int sh_rem, sh_eqbase, sh_cnt;
  int tid = threadIdx.x;
  if (tid == 0) { sh_prefix = 0u; sh_rem = KK; sh_eqbase = 0; sh_cnt = 0; }
  __syncthreads();
  // radix select: 4 passes of 8 bits on order-preserving key
  for (int pass = 0; pass < 4; ++pass) {
    int shift = 24 - pass * 8;
    unsigned himask = (pass == 0) ? 0u : (0xFFFFFFFFu << (shift + 8));
    hist[tid] = 0;
    __syncthreads();
    unsigned pfx = sh_prefix;
    for (int i = tid; i < NN; i += 256) {
      unsigned u = __float_as_uint(s[i]);
      unsigned key = (u & 0x80000000u) ? ~u : (u | 0x80000000u);
      if ((key & himask) == (pfx & himask))
        atomicAdd(&hist[(key >> shift) & 0xFFu], 1);
    }
    __syncthreads();
    if (tid == 0) {
      int rem = sh_rem, b = 255;
      for (; b > 0; --b) { if (hist[b] >= rem) break; rem -= hist[b]; }
      sh_prefix |= ((unsigned)b) << shift;
      sh_rem = rem;
    }
    __syncthreads();
  }
  unsigned T = sh_prefix;
  int need_eq = sh_rem;          // # keys == T to keep (lowest index first)
  int nchunk = (NN + 255) / 256;
  for (int c = 0; c < nchunk; ++c) {
    int i = c * 256 + tid;
    int eqf = 0, gtf = 0;
    if (i < NN) {
      unsigned u = __float_as_uint(s[i]);
      unsigned key = (u & 0x80000000u) ? ~u : (u | 0x80000000u);
      gtf = key > T; eqf = key == T;
    }
    scn[tid] = eqf;
    __syncthreads();
    for (int off = 1; off < 256; off <<= 1) {   // Hillis-Steele inclusive scan
      int t = (tid >= off) ? scn[tid - off] : 0;
      __syncthreads();
      scn[tid] += t;
      __syncthreads();
    }
    int excl = scn[tid] - eqf;
    int myrank = sh_eqbase + excl;
    if (gtf || (eqf && myrank < need_eq)) {
      int pos = atomicAdd(&sh_cnt, 1);
      if (pos < KK) {
        keep[(size_t)g * KK + pos] = g * NN + i;   // global node id
        if (g == 0) nidx0[i] = pos;
      }
    }
    __syncthreads();
    if (tid == 0) sh_eqbase += scn[255];
    __syncthreads();
  }
}

// ------------------------------------------------------- conv2 degrees
__global__ void k_deg2(const int* src, const int* dst, const int* nidx0,
                       float* deg2, int E) {
  int e = blockIdx.x * blockDim.x + threadIdx.x;
  if (e >= E) return;
  int ns = nidx0[src[e]], nd = nidx0[dst[e]];
  if (ns >= 0 && nd >= 0) atomicAdd(&deg2[nd], 1.f);
}

// -------------------------- conv2 dense part: fp32 WMMA 16x16x4, K=64
__global__ __launch_bounds__(256)
void k_conv2_wmma(const float* x, const float* xh1, const float* score,
                  const int* keep, const float* W1, const float* b1,
                  const float* W2, const float* b2, float* xw2, int* gbits) {
  __shared__ float sW1[FIN * H1D];
  __shared__ float sB1[H1D];
  __shared__ float sW2T[H2D * 66];        // W2 transposed, padded rows
  __shared__ float sA[TPB * 16 * 66];     // 8 tiles of 16x64 xp, padded rows
  int tid = threadIdx.x;
  int g = blockIdx.y;
  for (int i = tid; i < FIN * H1D; i += 256) sW1[i] = W1[i];
  if (tid < H1D) sB1[tid] = b1[tid];
  for (int i = tid; i < H1D * H2D; i += 256) {
    int k = i >> 5, n = i & 31;
    sW2T[n * 66 + k] = W2[i];
  }
  __syncthreads();
  // ---- stage xp tiles: xp = h1 * tanh(score); h1 recomputed for g>=1
  {
    int row = tid >> 1, half = tid & 1;   // 128 rows, 2 threads/row
    int tI = row >> 4, r = row & 15;
    int jj = (blockIdx.x * TPB + tI) * 16 + r;
    float* dstA = &sA[(tI * 16 + r) * 66 + half * 32];
    if (jj < KK) {
      int node = keep[(size_t)g * KK + jj];
      float th = tanhf(score[node]);
      if (g == 0) {
        const float* h = xh1 + (size_t)node * H1D + half * 32;
#pragma unroll
        for (int f = 0; f < 32; ++f) dstA[f] = h[f] * th;
      } else {
        const float* xr = x + (size_t)node * FIN;
        float x0 = xr[0], x1 = xr[1], x2 = xr[2], x3 = xr[3], x4 = xr[4], x5 = xr[5];
#pragma unroll
        for (int f = 0; f < 32; ++f) {
          int ff = half * 32 + f;
          float h = sB1[ff] + x0 * sW1[ff] + x1 * sW1[H1D + ff]
                            + x2 * sW1[2 * H1D + ff] + x3 * sW1[3 * H1D + ff]
                            + x4 * sW1[4 * H1D + ff] + x5 * sW1[5 * H1D + ff];
          dstA[f] = fmaxf(h, 0.f) * th;
        }
      }
    } else {
#pragma unroll
      for (int f = 0; f < 32; ++f) dstA[f] = 0.f;  // pad rows -> zeros
    }
  }
  __syncthreads();
  // ---- one wave per 16x32 output tile, 16 chained f32 16x16x4 WMMAs per half
  int wave = tid >> 5, lane = tid & 31;
  int lmod = lane & 15, khalf = lane >> 4;
  int jbase = (blockIdx.x * TPB + wave) * 16;
  const float* Aw  = &sA[wave * 16 * 66 + lmod * 66 + 2 * khalf];
  const float* Bc0 = &sW2T[lmod * 66 + 2 * khalf];
  const float* Bc1 = &sW2T[(16 + lmod) * 66 + 2 * khalf];
  v8f c0 = {}; v8f c1 = {};
#pragma unroll
  for (int kc = 0; kc < 16; ++kc) {
    int kb = kc * 4;
    v2f a   = { Aw[kb],  Aw[kb + 1]  };
    v2f b0  = { Bc0[kb], Bc0[kb + 1] };
    v2f b1v = { Bc1[kb], Bc1[kb + 1] };
    c0 = __builtin_amdgcn_wmma_f32_16x16x4_f32(false, a, false, b0,  (short)0, c0, false, false);
    c1 = __builtin_amdgcn_wmma_f32_16x16x4_f32(false, a, false, b1v, (short)0, c1, false, false);
  }
  int col0 = lmod, col1 = 16 + lmod;
  if (g == 0) {                 // graph 0: store raw xw2 for edge aggregation
#pragma unroll
    for (int r = 0; r < 8; ++r) {
      int j = jbase + r + 8 * khalf;
      if (j < KK) {
        xw2[(size_t)j * H2D + col0] = c0[r];
        xw2[(size_t)j * H2D + col1] = c1[r];
      }
    }
  } else {                      // graphs >= 1: h2 = relu(xw2 + b2); max-pool
    float bb0 = b2[col0], bb1 = b2[col1];
    float m0 = 0.f, m1 = 0.f;   // relu >= 0, so 0 is a safe identity
#pragma unroll
    for (int r = 0; r < 8; ++r) {
      int j = jbase + r + 8 * khalf;
      if (j < KK) {
        m0 = fmaxf(m0, fmaxf(c0[r] + bb0, 0.f));
        m1 = fmaxf(m1, fmaxf(c1[r] + bb1, 0.f));
      }
    }
    m0 = fmaxf(m0, __shfl_xor(m0, 16, 32));
    m1 = fmaxf(m1, __shfl_xor(m1, 16, 32));
    if (lane < 16) {            // non-negative floats: int-bit atomicMax is exact
      atomicMax(&gbits[g * H2D + col0], __float_as_int(m0));
      atomicMax(&gbits[g * H2D + col1], __float_as_int(m1));
    }
  }
}

// --------------------------------------------- conv2 graph-0 edge part
__global__ void k_conv2_edges(const int* src, const int* dst, const int* nidx0,
                              const float* dinv2, const float* xw2, float* acc2,
                              int E) {
  int idx = blockIdx.x * blockDim.x + threadIdx.x;
  if (idx >= E * H2D) return;
  int e = idx >> 5, f = idx & 31;
  int ns = nidx0[src[e]], nd = nidx0[dst[e]];
  if (ns < 0 || nd < 0) return;
  float w = dinv2[ns] * dinv2[nd];
  atomicAdd(&acc2[(size_t)nd * H2D + f], xw2[(size_t)ns * H2D + f] * w);
}

__global__ void k_g0max(const float* acc2, const float* xw2, const float* dinv2,
                        const float* b2, int* gbits) {
  int idx = blockIdx.x * blockDim.x + threadIdx.x;
  if (idx >= KK * H2D) return;
  int j = idx >> 5, f = idx & 31;
  float dv = dinv2[j];
  float v = fmaxf(acc2[idx] + xw2[idx] * dv * dv + b2[f], 0.f);
  atomicMax(&gbits[f], __float_as_int(v));   // graph 0 slots
}

// ------------------------------------------------------------- output
__global__ void k_final(const int* gbits, const float* Wf, const float* bf,
                        float* out, int Bsz) {
  int g = blockIdx.x * blockDim.x + threadIdx.x;
  if (g >= Bsz) return;
  float a = bf[0];
#pragma unroll
  for (int f = 0; f < H2D; ++f)
    a += __int_as_float(gbits[g * H2D + f]) * Wf[f];
  out[g] = 1.f / (1.f + expf(-a));
}

// =====================================================================
extern "C" void kernel_launch(void* const* d_in, const int* in_sizes, int n_in,
                              void* d_out, int out_size, void* d_ws, size_t ws_size,
                              hipStream_t stream) {
  (void)n_in; (void)out_size; (void)ws_size;
  const float* x  = (const float*)d_in[0];
  const int*   ei = (const int*)d_in[1];
  const float* W1 = (const float*)d_in[2];
  const float* b1 = (const float*)d_in[3];
  const float* Wp = (const float*)d_in[4];
  const float* bp = (const float*)d_in[5];
  const float* W2 = (const float*)d_in[6];
  const float* b2 = (const float*)d_in[7];
  const float* Wf = (const float*)d_in[8];
  const float* bf = (const float*)d_in[9];
  float* out = (float*)d_out;

  int Bsz = in_sizes[0] / (NN * FIN);
  int E   = in_sizes[1] / 2;
  const int* src = ei;
  const int* dst = ei + E;

  // workspace carve (total ~20 MB)
  float* ws = (float*)d_ws;
  size_t off = 0;
  auto alloc = [&](size_t n) { float* p = ws + off; off += (n + 63) & ~(size_t)63; return p; };
  float* dinv1 = alloc(NN);
  float* xh1   = alloc((size_t)NN * H1D);   // xw1, then h1 (graph 0) in-place
  float* acc1  = alloc((size_t)NN * H1D);
  float* sxw0  = alloc(NN);
  float* sacc  = alloc(NN);
  float* score = alloc((size_t)Bsz * NN);
  int*   keep  = (int*)alloc((size_t)Bsz * KK);
  int*   nidx0 = (int*)alloc(NN);
  float* dinv2 = alloc(KK);
  float* xw2   = alloc((size_t)KK * H2D);
  float* acc2  = alloc((size_t)KK * H2D);
  int*   gbits = (int*)alloc((size_t)Bsz * H2D);

  const int T = 256;
  auto nb = [&](long long n) { return (unsigned)((n + T - 1) / T); };
  long long Ntot = (long long)Bsz * NN;
  long long ninit = (long long)NN * H1D;
  if ((long long)Bsz * KK > ninit) ninit = (long long)Bsz * KK;

  k_init<<<nb(ninit), T, 0, stream>>>(dinv1, acc1, sacc, nidx0, dinv2, acc2,
                                      gbits, keep, Bsz, ninit);
  // conv1 (graph 0 edges) + h1
  k_deg1<<<nb(E), T, 0, stream>>>(dst, dinv1, E);
  k_rsqrt<<<nb(NN), T, 0, stream>>>(dinv1, NN);
  k_xw1<<<nb((long long)NN * H1D), T, 0, stream>>>(x, W1, xh1);
  k_conv1_edges<<<nb((long long)E * H1D), T, 0, stream>>>(src, dst, dinv1, xh1, acc1, E);
  k_h1fin<<<nb((long long)NN * H1D), T, 0, stream>>>(xh1, acc1, dinv1, b1);
  // SAGPool score
  k_score<<<nb(Ntot), T, 0, stream>>>(x, xh1, W1, b1, Wp, bp, sxw0, score, Ntot);
  k_score_edges<<<nb(E), T, 0, stream>>>(src, dst, dinv1, sxw0, sacc, E);
  k_score0<<<nb(NN), T, 0, stream>>>(sacc, sxw0, dinv1, bp, score);
  // exact per-graph top-K (set semantics)
  k_topk<<<Bsz, T, 0, stream>>>(score, keep, nidx0);
  // conv2 degrees (graph 0, filtered edges)
  k_deg2<<<nb(E), T, 0, stream>>>(src, dst, nidx0, dinv2, E);
  k_rsqrt<<<nb(KK), T, 0, stream>>>(dinv2, KK);
  // conv2 dense part via fp32 WMMA + fused max-pool (graphs >= 1)
  k_conv2_wmma<<<dim3(GXW, Bsz), T, 0, stream>>>(x, xh1, score, keep, W1, b1,
                                                 W2, b2, xw2, gbits);
  // conv2 graph-0 aggregation + finalize + max-pool
  k_conv2_edges<<<nb((long long)E * H2D), T, 0, stream>>>(src, dst, nidx0, dinv2,
                                                          xw2, acc2, E);
  k_g0max<<<nb((long long)KK * H2D), T, 0, stream>>>(acc2, xw2, dinv2, b2, gbits);
  // readout
  k_final<<<nb(Bsz), T, 0, stream>>>(gbits, Wf, bf, out, Bsz);
}